// RetinaNet_11897059410439
// MI455X (gfx1250) — compile-verified
//
#include <hip/hip_runtime.h>
#include <stdint.h>

// ---------------------------------------------------------------------------
// RetinaNet loss (fused): anchor assignment (IoU max/argmax vs M GT boxes),
// sigmoid focal loss (alpha=.25, gamma=2), smooth-L1 on positives, global
// normalized sums -> scalar.
// Memory-bound: ~29.5MB cls_logits + 1.2MB anchors + sparse delta lines
// (~31MB @ 23.3TB/s ~ 1.4us). Design: single streaming pass, GT staged to
// LDS via CDNA5 async global->LDS DMA, b128 NT loads for the big stream
// hoisted above the IoU loop, WMMA-assisted wave reduction, deterministic
// two-stage final reduce (no float atomics -> bitwise-stable replays).
// ---------------------------------------------------------------------------

typedef float v2f __attribute__((ext_vector_type(2)));
typedef float v4f __attribute__((ext_vector_type(4)));
typedef float v8f __attribute__((ext_vector_type(8)));

#ifndef MAX_M
#define MAX_M 64
#endif

#ifdef __has_builtin
#if __has_builtin(__builtin_amdgcn_wmma_f32_16x16x4_f32)
#define RETINA_USE_WMMA 1
#endif
#endif

// Full 32-lane sum, broadcast to all lanes.
// WMMA stage: A = ones(16x4), B vgpr0 holds the 32 lane values as two K-rows
// of B(4x16); every element of D is the column sum over K, so D[0] in lane L
// is v[L&15] + v[(L&15)+16] regardless of the exact K<->VGPR-half mapping.
// The remaining 16->1 fold is a 4-step xor tree.
__device__ __forceinline__ float wave_sum32(float v) {
#ifdef RETINA_USE_WMMA
  v2f a;  a[0] = 1.0f; a[1] = 1.0f;
  v2f bm; bm[0] = v;   bm[1] = 0.0f;
  v8f c = {};
  c = __builtin_amdgcn_wmma_f32_16x16x4_f32(
      /*neg_a=*/false, a, /*neg_b=*/false, bm,
      /*c_mod=*/(short)0, c, /*reuse_a=*/false, /*reuse_b=*/false);
  float s = c[0];
#else
  float s = v + __shfl_xor(v, 16, 32);
#endif
  s += __shfl_xor(s, 8, 32);
  s += __shfl_xor(s, 4, 32);
  s += __shfl_xor(s, 2, 32);
  s += __shfl_xor(s, 1, 32);
  return s;
}

// Numerically stable focal term for one logit x with binary target t.
__device__ __forceinline__ float focal_term(float x, float t) {
  const float l1p = log1pf(expf(-fabsf(x)));
  const float lsp = fminf(x, 0.0f) - l1p;    // log sigmoid(x)
  const float lsn = fminf(-x, 0.0f) - l1p;   // log sigmoid(-x)
  const float ce  = -(t * lsp + (1.0f - t) * lsn);
  const float p   = 1.0f / (1.0f + expf(-x));
  const float pt  = t * p + (1.0f - t) * (1.0f - p);
  const float at  = t * 0.25f + (1.0f - t) * 0.75f;
  const float om  = 1.0f - pt;
  return at * om * om * ce;
}

// CT > 0: compile-time class count (vectorized b128 logit loads, unrolled).
// CT == 0: runtime class count fallback (scalar streaming loads).
template <int CT>
__global__ void __launch_bounds__(256)
retina_partial(const float* __restrict__ cls_logits,       // [B,N,C]
               const float* __restrict__ box_deltas,       // [B,N,4]
               const float* __restrict__ anchors,          // [N,4]
               const float* __restrict__ gt_boxes,         // [B,M,4]
               const int* __restrict__ gt_labels,          // [B,M]
               const unsigned char* __restrict__ gt_valid, // [B,M] (bool)
               float* __restrict__ partials,               // [gridY*gridX, 4]
               int N, int M, int Crt) {
  const int C = (CT > 0) ? CT : Crt;

  __shared__ __align__(16) float s_box[MAX_M][4];
  __shared__ int                 s_lab[MAX_M];
  __shared__ unsigned char       s_val[MAX_M];
  __shared__ unsigned long long  s_valmask;
  __shared__ float               s_red[8][4];

  const int tid = threadIdx.x;
  const int b   = blockIdx.y;
  const int n   = blockIdx.x * blockDim.x + tid;

  // --- stage this batch's GT into LDS; boxes via CDNA5 async global->LDS ---
  if (tid < 32) {
    for (int m = tid; m < M && m < MAX_M; m += 32) {
      uint32_t lds_off = (uint32_t)(uintptr_t)(&s_box[m][0]);
      uint64_t gaddr   = (uint64_t)(uintptr_t)(gt_boxes + ((size_t)b * M + m) * 4);
      asm volatile("global_load_async_to_lds_b128 %0, %1, off"
                   :: "v"(lds_off), "v"(gaddr) : "memory");
      s_lab[m] = gt_labels[b * M + m];
      s_val[m] = gt_valid[b * M + m];
    }
    asm volatile("s_wait_asynccnt 0" ::: "memory");
  }
  __syncthreads();
  if (tid == 0) {
    unsigned long long mk = 0;
    for (int m = 0; m < M && m < MAX_M; ++m)
      if (s_val[m]) mk |= 1ull << m;
    s_valmask = mk;
  }
  __syncthreads();

  float acc_focal = 0.0f, acc_valid = 0.0f, acc_sl1 = 0.0f, acc_pos = 0.0f;

  if (n < N) {
    const v4f a4 = *(const v4f*)(anchors + (size_t)n * 4);
    const float ax1 = a4[0], ay1 = a4[1], ax2 = a4[2], ay2 = a4[3];
    const float aw = ax2 - ax1, ah = ay2 - ay1;
    const float area_a = aw * ah;

    const float* cls_row = cls_logits + ((size_t)b * N + n) * (size_t)C;

    // Hoist the dominant 29.5MB stream above the IoU loop (latency hiding);
    // b128 non-temporal loads when C is compile-time and divisible by 4.
    v4f xv[(CT > 0) ? (CT + 3) / 4 : 1];
    if (CT > 0 && (CT % 4) == 0) {
#pragma unroll
      for (int q = 0; q < CT / 4; ++q)
        xv[q] = __builtin_nontemporal_load((const v4f*)cls_row + q);
    }

    const unsigned long long vm = s_valmask;

    // --- IoU max / argmax over M GT boxes (first-max tie break = argmax) ---
    float best = -2.0f; int bidx = 0;
    for (int m = 0; m < M && m < MAX_M; ++m) {
      const v4f g = *(const v4f*)&s_box[m][0];   // LDS broadcast b128
      const float gx1 = g[0], gy1 = g[1], gx2 = g[2], gy2 = g[3];
      float iw = fminf(ax2, gx2) - fmaxf(ax1, gx1); iw = fmaxf(iw, 0.0f);
      float ih = fminf(ay2, gy2) - fmaxf(ay1, gy1); ih = fmaxf(ih, 0.0f);
      const float inter  = iw * ih;
      const float area_g = (gx2 - gx1) * (gy2 - gy1);
      float iou = inter / (area_a + area_g - inter);
      iou = ((vm >> m) & 1ull) ? iou : -1.0f;
      if (iou > best) { best = iou; bidx = m; }
    }
    const bool pos   = best >= 0.5f;
    const bool valid = pos || (best < 0.4f);
    const int  mlab  = s_lab[bidx];

    // --- sigmoid focal loss over C classes ---
    float fsum = 0.0f;
    if (CT > 0 && (CT % 4) == 0) {
#pragma unroll
      for (int c = 0; c < CT; ++c) {
        const float x = xv[c >> 2][c & 3];
        const float t = (pos && c == mlab) ? 1.0f : 0.0f;
        fsum += focal_term(x, t);
      }
    } else {
      for (int c = 0; c < C; ++c) {
        const float x = __builtin_nontemporal_load(cls_row + c);
        const float t = (pos && c == mlab) ? 1.0f : 0.0f;
        fsum += focal_term(x, t);
      }
    }
    if (valid) { acc_focal = fsum; acc_valid = 1.0f; }

    // --- smooth-L1 on positives only: deltas are loaded ONLY here, so the
    //     9.8MB box_deltas tensor contributes just the sparse positive lines.
    if (pos) {
      const float gx1 = s_box[bidx][0], gy1 = s_box[bidx][1];
      const float gx2 = s_box[bidx][2], gy2 = s_box[bidx][3];
      const float gw = gx2 - gx1, gh = gy2 - gy1;
      const float gcx = gx1 + 0.5f * gw, gcy = gy1 + 0.5f * gh;
      const float acx = ax1 + 0.5f * aw, acy = ay1 + 0.5f * ah;
      const float* del_row = box_deltas + ((size_t)b * N + n) * 4;
      const v4f d4 = __builtin_nontemporal_load((const v4f*)del_row);
      const float tt[4] = { (gcx - acx) / aw, (gcy - acy) / ah,
                            logf(gw / aw),    logf(gh / ah) };
      float s = 0.0f;
#pragma unroll
      for (int k = 0; k < 4; ++k) {
        const float d = fabsf(d4[k] - tt[k]);
        s += (d < 1.0f) ? 0.5f * d * d : d - 0.5f;
      }
      acc_sl1 = s; acc_pos = 1.0f;
    }
  }

  // --- block reduction: WMMA wave fold, then LDS across the 8 waves ---
  const float r0 = wave_sum32(acc_focal);
  const float r1 = wave_sum32(acc_valid);
  const float r2 = wave_sum32(acc_sl1);
  const float r3 = wave_sum32(acc_pos);
  const int wave = tid >> 5, lane = tid & 31;
  if (lane == 0) {
    s_red[wave][0] = r0; s_red[wave][1] = r1;
    s_red[wave][2] = r2; s_red[wave][3] = r3;
  }
  __syncthreads();
  if (tid == 0) {
    float o0 = 0, o1 = 0, o2 = 0, o3 = 0;
    const int nw = (int)((blockDim.x + 31) >> 5);
    for (int w = 0; w < nw; ++w) {
      o0 += s_red[w][0]; o1 += s_red[w][1];
      o2 += s_red[w][2]; o3 += s_red[w][3];
    }
    v4f o; o[0] = o0; o[1] = o1; o[2] = o2; o[3] = o3;
    *(v4f*)(partials + ((size_t)blockIdx.y * gridDim.x + blockIdx.x) * 4) = o;
  }
}

__global__ void __launch_bounds__(256)
retina_finalize(const float* __restrict__ partials, int nPart,
                float* __restrict__ out, int C) {
  __shared__ float s_red[8][4];
  const int tid = threadIdx.x;
  float a0 = 0, a1 = 0, a2 = 0, a3 = 0;
  for (int i = tid; i < nPart; i += (int)blockDim.x) {
    const v4f p = *(const v4f*)(partials + (size_t)i * 4);
    a0 += p[0]; a1 += p[1]; a2 += p[2]; a3 += p[3];
  }
  a0 = wave_sum32(a0); a1 = wave_sum32(a1);
  a2 = wave_sum32(a2); a3 = wave_sum32(a3);
  const int wave = tid >> 5, lane = tid & 31;
  if (lane == 0) {
    s_red[wave][0] = a0; s_red[wave][1] = a1;
    s_red[wave][2] = a2; s_red[wave][3] = a3;
  }
  __syncthreads();
  if (tid == 0) {
    float o0 = 0, o1 = 0, o2 = 0, o3 = 0;
    const int nw = (int)((blockDim.x + 31) >> 5);
    for (int w = 0; w < nw; ++w) {
      o0 += s_red[w][0]; o1 += s_red[w][1];
      o2 += s_red[w][2]; o3 += s_red[w][3];
    }
    const float cls_loss = o0 / fmaxf(o1 * (float)C, 1.0f);
    const float box_loss = o2 / fmaxf(o3 * 4.0f, 1.0f);
    out[0] = cls_loss + box_loss;
  }
}

extern "C" void kernel_launch(void* const* d_in, const int* in_sizes, int n_in,
                              void* d_out, int out_size, void* d_ws, size_t ws_size,
                              hipStream_t stream) {
  const float*         cls_logits = (const float*)d_in[0];
  const float*         box_deltas = (const float*)d_in[1];
  const float*         anchors    = (const float*)d_in[2];
  const float*         gt_boxes   = (const float*)d_in[3];
  const int*           gt_labels  = (const int*)d_in[4];
  const unsigned char* gt_valid   = (const unsigned char*)d_in[5];

  const int N = in_sizes[2] / 4;                 // anchors [N,4]
  if (N <= 0) return;
  const int B = in_sizes[1] / (4 * N);           // box_deltas [B,N,4]
  if (B <= 0) return;
  const int M = in_sizes[4] / B;                 // gt_labels [B,M]
  const int C = (int)((long long)in_sizes[0] / ((long long)B * N)); // cls [B,N,C]

  const int threads = 256;
  const int gx = (N + threads - 1) / threads;
  dim3 grid(gx, B);

  float* partials = (float*)d_ws;                // gx*B*4 floats (~38 KB here)

  if (C == 12) {
    retina_partial<12><<<grid, threads, 0, stream>>>(
        cls_logits, box_deltas, anchors, gt_boxes, gt_labels, gt_valid,
        partials, N, M, C);
  } else {
    retina_partial<0><<<grid, threads, 0, stream>>>(
        cls_logits, box_deltas, anchors, gt_boxes, gt_labels, gt_valid,
        partials, N, M, C);
  }

  retina_finalize<<<1, threads, 0, stream>>>(partials, gx * B, (float*)d_out, C);
}